// attention_module_multi_head_37495064494465
// MI455X (gfx1250) — compile-verified
//
#include <hip/hip_runtime.h>
#include <hip/hip_bf16.h>

typedef __attribute__((ext_vector_type(2))) float v2f;
typedef __attribute__((ext_vector_type(8))) float v8f;
typedef __attribute__((ext_vector_type(4))) int   v4i;

#define WMMA_F32(a, b, c) \
    __builtin_amdgcn_wmma_f32_16x16x4_f32(false, (a), false, (b), (short)0, (c), false, false)

#define NROI 1024
#define DIM  1024
#define PDIM 64
#define GRP  16
#define DK   64

// ---------------------------------------------------------------------------
// CDNA5 async global->LDS copy (ASYNCcnt path), with synchronous fallback so
// the file compiles on toolchains lacking the builtin.
// ---------------------------------------------------------------------------
#if defined(__HIP_DEVICE_COMPILE__) && __has_builtin(__builtin_amdgcn_global_load_async_to_lds_b128)
#define HAVE_ASYNC_LDS 1
#else
#define HAVE_ASYNC_LDS 0
#endif

__device__ __forceinline__ void async_copy16(const void* g, void* l)
{
#if HAVE_ASYNC_LDS
    typedef __attribute__((address_space(1))) v4i* g4p;
    typedef __attribute__((address_space(3))) v4i* l4p;
    __builtin_amdgcn_global_load_async_to_lds_b128(
        (g4p)(uintptr_t)g, (l4p)(uintptr_t)l, 0, 0);
#else
    *(float4*)l = *(const float4*)g;
#endif
}

template <int N>
__device__ __forceinline__ void wait_async()
{
#if HAVE_ASYNC_LDS
#if __has_builtin(__builtin_amdgcn_s_wait_asynccnt)
    __builtin_amdgcn_s_wait_asynccnt((unsigned short)N);
#else
    if (N == 0) asm volatile("s_wait_asynccnt 0x0" ::: "memory");
    else        asm volatile("s_wait_asynccnt 0x2" ::: "memory");
#endif
#endif
}

// ---------------------------------------------------------------------------
// Generic fp32 WMMA GEMM: C[M,N] = A[M,K] @ B[K,N] (+bias), 64x64 tile / WG,
// 8 waves (wave32), 16x32 sub-tile per wave. Double-buffered LDS with async
// global->LDS copies so tile t+1 streams in under tile t's WMMAs.
// transC=1 stores C transposed: C[n*ldc + m] (used for k^T layout).
// blockIdx.z batches B and C by strideB/strideC (A shared across batch).
// ---------------------------------------------------------------------------
__global__ __launch_bounds__(256)
void gemm64x64(const float* __restrict__ A, const float* __restrict__ B,
               const float* __restrict__ bias, float* __restrict__ C,
               int lda, int ldb, int ldc, int K,
               long strideB, long strideC, int transC)
{
    __shared__ float As[2][64][20];   // 64x16 tile, pad 20 (conflict-free A-frag reads)
    __shared__ float Bs[2][16][64];   // 16x64 tile

    const int tid  = threadIdx.x;
    const int lane = tid & 31;
    const int w    = tid >> 5;
    const int m0   = blockIdx.y * 64;
    const int n0   = blockIdx.x * 64;
    B += (long)blockIdx.z * strideB;
    C += (long)blockIdx.z * strideC;

    const int wm = (w & 3) * 16;      // wave row base within 64
    const int wn = (w >> 2) * 32;     // wave col base within 64
    const int lm = lane & 15;
    const int lk = (lane >> 4) * 2;   // K offset pair per half-wave

    // per-thread copy slots (one 16B A chunk + one 16B B chunk per tile)
    const int ra = tid >> 2, ca = (tid & 3) * 4;
    const int rb = tid >> 4, cb = (tid & 15) * 4;
    const float* gA = A + (long)(m0 + ra) * lda + ca;
    const float* gB = B + (long)rb * ldb + n0 + cb;

    v8f acc0 = {0.f,0.f,0.f,0.f,0.f,0.f,0.f,0.f};
    v8f acc1 = {0.f,0.f,0.f,0.f,0.f,0.f,0.f,0.f};

    const int ntiles = K >> 4;
    // prologue: stream tile 0 into buffer 0
    async_copy16(gA, &As[0][ra][ca]);
    async_copy16(gB, &Bs[0][rb][cb]);

    for (int t = 0; t < ntiles; ++t) {
        const int buf = t & 1;
        if (t + 1 < ntiles) {            // stream next tile into other buffer
            const int kk = (t + 1) << 4;
            async_copy16(gA + kk,               &As[buf ^ 1][ra][ca]);
            async_copy16(gB + (long)kk * ldb,   &Bs[buf ^ 1][rb][cb]);
            wait_async<2>();             // tile t landed (async loads retire in order)
        } else {
            wait_async<0>();
        }
        __syncthreads();

        #pragma unroll
        for (int k2 = 0; k2 < 16; k2 += 4) {
            v2f a;  a.x  = As[buf][wm + lm][k2 + lk];       a.y  = As[buf][wm + lm][k2 + lk + 1];
            v2f b0; b0.x = Bs[buf][k2 + lk][wn + lm];       b0.y = Bs[buf][k2 + lk + 1][wn + lm];
            v2f b1; b1.x = Bs[buf][k2 + lk][wn + 16 + lm];  b1.y = Bs[buf][k2 + lk + 1][wn + 16 + lm];
            acc0 = WMMA_F32(a, b0, acc0);
            acc1 = WMMA_F32(a, b1, acc1);
        }
        __syncthreads();                 // buffer free for reuse at t+2
    }

    #pragma unroll
    for (int v = 0; v < 8; ++v) {
        int r  = m0 + wm + v + (lane >> 4) * 8;
        int c0 = n0 + wn + lm;
        float o0 = acc0[v], o1 = acc1[v];
        if (bias) { o0 += bias[c0]; o1 += bias[c0 + 16]; }
        if (!transC) {
            C[(long)r * ldc + c0]      = o0;
            C[(long)r * ldc + c0 + 16] = o1;
        } else {
            C[(long)c0 * ldc + r]        = o0;
            C[(long)(c0 + 16) * ldc + r] = o1;
        }
    }
}

// ---------------------------------------------------------------------------
// Positional bias: aff[i][g][j] = tanh(pe[i,j,:] @ W1 + b1) @ W2 + b2
// One WG handles (i fixed, 64 consecutive j). fc1 = 64x64x64 WMMA, tanh,
// fc2 = 64x64x16 WMMA, scattered store into the [i][g][j] layout.
// This kernel streams the 256 MB position embedding: the HBM-bound stage.
// ---------------------------------------------------------------------------
__global__ __launch_bounds__(256)
void posbias(const float* __restrict__ pe, const float* __restrict__ W1,
             const float* __restrict__ b1, const float* __restrict__ W2,
             const float* __restrict__ b2, float* __restrict__ aff)
{
    __shared__ float Xs[64][68];
    __shared__ float W1s[64][64];
    __shared__ float C1s[64][68];
    __shared__ float W2s[64][16];

    const int tid  = threadIdx.x;
    const int lane = tid & 31;
    const int w    = tid >> 5;
    const int i    = blockIdx.y;
    const int j0   = blockIdx.x * 64;
    const float* X = pe + ((long)i * NROI + j0) * PDIM;

    #pragma unroll
    for (int it = 0; it < 4; ++it) {
        int idx = tid + it * 256;
        int r = idx >> 4, c4 = (idx & 15) * 4;
        async_copy16(X + (long)r * PDIM + c4, &Xs[r][c4]);
        async_copy16(W1 + r * PDIM + c4,      &W1s[r][c4]);
    }
    {   int r = tid >> 2, c4 = (tid & 3) * 4;
        async_copy16(W2 + r * GRP + c4, &W2s[r][c4]);
    }
    wait_async<0>();
    __syncthreads();

    const int lm = lane & 15;
    const int lk = (lane >> 4) * 2;
    const int wm = (w & 3) * 16, wn = (w >> 2) * 32;

    v8f c0 = {0.f,0.f,0.f,0.f,0.f,0.f,0.f,0.f};
    v8f c1 = {0.f,0.f,0.f,0.f,0.f,0.f,0.f,0.f};
    #pragma unroll
    for (int kk = 0; kk < 64; kk += 4) {
        v2f a;  a.x  = Xs[wm + lm][kk + lk];        a.y  = Xs[wm + lm][kk + lk + 1];
        v2f b0; b0.x = W1s[kk + lk][wn + lm];       b0.y = W1s[kk + lk + 1][wn + lm];
        v2f b1; b1.x = W1s[kk + lk][wn + 16 + lm];  b1.y = W1s[kk + lk + 1][wn + 16 + lm];
        c0 = WMMA_F32(a, b0, c0);
        c1 = WMMA_F32(a, b1, c1);
    }
    #pragma unroll
    for (int v = 0; v < 8; ++v) {
        int r = wm + v + (lane >> 4) * 8;
        int n = wn + lm;
        C1s[r][n]      = tanhf(c0[v] + b1[n]);
        C1s[r][n + 16] = tanhf(c1[v] + b1[n + 16]);
    }
    __syncthreads();

    if (w < 4) {   // wave-uniform: EXEC stays all-ones inside
        const int wm2 = w * 16;
        v8f c2 = {0.f,0.f,0.f,0.f,0.f,0.f,0.f,0.f};
        #pragma unroll
        for (int kk = 0; kk < 64; kk += 4) {
            v2f a; a.x = C1s[wm2 + lm][kk + lk]; a.y = C1s[wm2 + lm][kk + lk + 1];
            v2f b; b.x = W2s[kk + lk][lm];       b.y = W2s[kk + lk + 1][lm];
            c2 = WMMA_F32(a, b, c2);
        }
        #pragma unroll
        for (int v = 0; v < 8; ++v) {
            int r = wm2 + v + (lane >> 4) * 8;     // local j
            int g = lm;
            aff[(long)i * (GRP * NROI) + (long)g * NROI + (j0 + r)] = c2[v] + b2[g];
        }
    }
}

// ---------------------------------------------------------------------------
// Attention: per (g, 16-row i-tile): S = (Q_g K_g^T) * aff * 0.125, softmax
// over 1024 keys, then P @ VW_g (VW = V @ W_out precomputed) + b_out.
// 8 waves split the 1024-key dimension. kt is K^T [DIM][NROI] so B-fragment
// loads are lane-coalesced.
// ---------------------------------------------------------------------------
__global__ __launch_bounds__(256)
void attn(const float* __restrict__ qbuf, const float* __restrict__ kt,
          const float* __restrict__ aff, const float* __restrict__ vw,
          const float* __restrict__ b_out, float* __restrict__ out)
{
    __shared__ float Qs[16][68];
    __shared__ float S[16][1028];       // pad 1028: stride mod 64 == 4 -> conflict-free A reads
    __shared__ float Pp[8][16][64];
    __shared__ float red[16][16];
    __shared__ float rowstat[16];

    const int tid  = threadIdx.x;
    const int lane = tid & 31;
    const int w    = tid >> 5;
    const int g    = blockIdx.x;
    const int i0   = blockIdx.y * 16;

    {   int r = tid >> 4, c4 = (tid & 15) * 4;
        async_copy16(qbuf + (long)(i0 + r) * DIM + g * DK + c4, &Qs[r][c4]);
    }
    wait_async<0>();
    __syncthreads();

    const int lm = lane & 15;
    const int lk = (lane >> 4) * 2;
    const int jb = w * 128;

    // ---- stage 1: logits for this wave's 128 keys ----
    #pragma unroll 1
    for (int tj = 0; tj < 8; ++tj) {
        const int j0t = jb + tj * 16;
        v8f acc = {0.f,0.f,0.f,0.f,0.f,0.f,0.f,0.f};
        #pragma unroll
        for (int kk = 0; kk < DK; kk += 4) {
            v2f a = *(const v2f*)&Qs[lm][kk + lk];
            const float* kp = kt + (long)(g * DK + kk + lk) * NROI + j0t + lm; // coalesced
            v2f b; b.x = kp[0]; b.y = kp[NROI];
            acc = WMMA_F32(a, b, acc);
        }
        #pragma unroll
        for (int v = 0; v < 8; ++v) {
            int il = v + (lane >> 4) * 8;
            int j  = j0t + lm;
            float aw = aff[(long)(i0 + il) * (GRP * NROI) + (long)g * NROI + j];
            S[il][j] = aw * acc[v] * 0.125f;   // 1/sqrt(64)
        }
    }
    __syncthreads();

    // ---- softmax over 1024 keys per row: 16 threads per row ----
    {
        const int row = tid >> 4, sub = tid & 15;
        float m = -3.402823466e38f;
        for (int kx = sub; kx < NROI; kx += 16) m = fmaxf(m, S[row][kx]);
        red[row][sub] = m;
        __syncthreads();
        if (sub == 0) {
            float mm = red[row][0];
            #pragma unroll
            for (int t = 1; t < 16; ++t) mm = fmaxf(mm, red[row][t]);
            rowstat[row] = mm;
        }
        __syncthreads();
        const float mm = rowstat[row];
        float s = 0.f;
        for (int kx = sub; kx < NROI; kx += 16) {
            float e = __expf(S[row][kx] - mm);
            S[row][kx] = e;
            s += e;
        }
        red[row][sub] = s;
        __syncthreads();
        if (sub == 0) {
            float ss = 0.f;
            #pragma unroll
            for (int t = 0; t < 16; ++t) ss += red[row][t];
            rowstat[row] = 1.f / ss;
        }
        __syncthreads();
        const float inv = rowstat[row];
        for (int kx = sub; kx < NROI; kx += 16) S[row][kx] *= inv;
    }
    __syncthreads();

    // ---- stage 2: P(16x128 per wave) @ VW_g(128x64 slice) -> partial 16x64 ----
    v8f o[4] = { {0.f,0.f,0.f,0.f,0.f,0.f,0.f,0.f}, {0.f,0.f,0.f,0.f,0.f,0.f,0.f,0.f},
                 {0.f,0.f,0.f,0.f,0.f,0.f,0.f,0.f}, {0.f,0.f,0.f,0.f,0.f,0.f,0.f,0.f} };
    #pragma unroll 1
    for (int kk = 0; kk < 128; kk += 4) {
        const int kbase = jb + kk + lk;
        v2f a = *(const v2f*)&S[lm][kbase];
        const float* vp = vw + (long)g * (NROI * DK) + (long)kbase * DK + lm;
        #pragma unroll
        for (int nt = 0; nt < 4; ++nt) {
            v2f b; b.x = vp[nt * 16]; b.y = vp[DK + nt * 16];
            o[nt] = WMMA_F32(a, b, o[nt]);
        }
    }
    #pragma unroll
    for (int nt = 0; nt < 4; ++nt) {
        #pragma unroll
        for (int v = 0; v < 8; ++v) {
            int il = v + (lane >> 4) * 8;
            Pp[w][il][nt * 16 + lm] = o[nt][v];
        }
    }
    __syncthreads();

    // ---- cross-wave reduce + bias + store ----
    #pragma unroll
    for (int it = 0; it < 4; ++it) {
        int e = tid + it * 256;
        int r = e >> 6, oc = e & 63;
        float s = 0.f;
        #pragma unroll
        for (int ww = 0; ww < 8; ++ww) s += Pp[ww][r][oc];
        out[(long)(i0 + r) * DIM + g * DK + oc] = s + b_out[g * DK + oc];
    }
}

// ---------------------------------------------------------------------------
extern "C" void kernel_launch(void* const* d_in, const int* in_sizes, int n_in,
                              void* d_out, int out_size, void* d_ws, size_t ws_size,
                              hipStream_t stream)
{
    const float* roi   = (const float*)d_in[0];
    const float* pe    = (const float*)d_in[1];
    const float* W_fc1 = (const float*)d_in[2];
    const float* b_fc1 = (const float*)d_in[3];
    const float* W_fc2 = (const float*)d_in[4];
    const float* b_fc2 = (const float*)d_in[5];
    const float* W_q   = (const float*)d_in[6];
    const float* b_q   = (const float*)d_in[7];
    const float* W_k   = (const float*)d_in[8];
    const float* b_k   = (const float*)d_in[9];
    const float* W_out = (const float*)d_in[10];
    const float* b_out = (const float*)d_in[11];
    float* out = (float*)d_out;

    float* ws   = (float*)d_ws;
    float* qb   = ws;                               // q  [1024,1024]
    float* ktb  = qb + (long)NROI * DIM;            // k^T [1024(D),1024(N)]
    float* vwb  = ktb + (long)DIM * NROI;           // VW [16,1024,64]
    float* affb = vwb + (long)GRP * NROI * DK;      // aff [1024,16,1024]
    (void)in_sizes; (void)n_in; (void)out_size; (void)ws_size;

    dim3 blk(256);

    // q = roi @ W_q + b_q ; kt = (roi @ W_k + b_k)^T
    gemm64x64<<<dim3(DIM / 64, NROI / 64, 1), blk, 0, stream>>>(
        roi, W_q, b_q, qb, DIM, DIM, DIM, DIM, 0, 0, 0);
    gemm64x64<<<dim3(DIM / 64, NROI / 64, 1), blk, 0, stream>>>(
        roi, W_k, b_k, ktb, DIM, DIM, NROI, DIM, 0, 0, 1);

    // VW[g] = V @ W_out[g]   (bias folded into attn kernel; softmax sums to 1)
    gemm64x64<<<dim3(1, NROI / 64, GRP), blk, 0, stream>>>(
        roi, W_out, nullptr, vwb, DIM, DK, DK, DIM,
        (long)DIM * DK, (long)NROI * DK, 0);

    // aff[i][g][j] = tanh(pe@W1+b1)@W2+b2  (streams the 256 MB PE tensor)
    posbias<<<dim3(NROI / 64, NROI, 1), blk, 0, stream>>>(
        pe, W_fc1, b_fc1, W_fc2, b_fc2, affb);

    // fused logits * aff -> softmax -> @ VW + b_out
    attn<<<dim3(GRP, NROI / 16, 1), blk, 0, stream>>>(
        qb, ktb, affb, vwb, b_out, out);
}